// SimpleSLM_88656714925063
// MI455X (gfx1250) — compile-verified
//
#include <hip/hip_runtime.h>
#include <hip/hip_bf16.h>

#define DEVI __device__ __forceinline__

typedef __attribute__((ext_vector_type(2))) float v2f;
typedef __attribute__((ext_vector_type(8))) float v8f;

namespace {
constexpr int NV = 96, NC = 64, NH = 4, NHS = 16, NL = 4, NT = 64, NB = 4096;
constexpr int XLD    = 68;   // row stride for 64-wide LDS tiles (bank-conflict free)
constexpr int QLD    = 17;   // row stride for per-head 16-wide q/k/v
constexpr int HID_LD = 260;  // row stride for MLP hidden [64][256]
constexpr int LLD    = 100;  // row stride for logits staging [64][96]
constexpr float EPS  = 1e-5f;

// 16x16 f32 tile: acc += A[16xK](LDS,row-major,lda) * B[Kx16](element (k,n) = Bp[k*bks+n*bns])
// Uses V_WMMA_F32_16X16X4_F32 in K/4 steps. Frag layout per CDNA5 ISA 7.12.2:
// component j of A/B frag holds K = 2*(lane>=16) + j.
DEVI v8f mm16(const float* A, int lda, const float* Bp, int bks, int bns, int K, v8f acc) {
  const int lane = threadIdx.x & 31;
  const int half = lane >> 4;
  const int l16  = lane & 15;
  const float* Arow = A + l16 * lda;
  for (int k = 0; k < K; k += 4) {
    const int ka = k + 2 * half;
    v2f a, b;
    a.x = Arow[ka];
    a.y = Arow[ka + 1];
    b.x = Bp[ka * bks + l16 * bns];
    b.y = Bp[(ka + 1) * bks + l16 * bns];
    acc = __builtin_amdgcn_wmma_f32_16x16x4_f32(false, a, false, b, (short)0, acc,
                                                false, false);
  }
  return acc;
}

// D tile (m,n): vgpr = m&7, lane = 16*(m>>3) + n
DEVI void store_tile(float* D, int ldd, v8f acc) {
  const int lane = threadIdx.x & 31;
  const int half = lane >> 4;
  const int l16  = lane & 15;
#pragma unroll
  for (int i = 0; i < 8; i++) D[(i + 8 * half) * ldd + l16] = acc[i];
}

DEVI void ln_rows(const float* src, float* dst, const float* g, const float* b, int tid) {
  if (tid < NT) {
    const float* r = src + tid * XLD;
    float m = 0.f;
    for (int c = 0; c < NC; c++) m += r[c];
    m *= (1.f / NC);
    float var = 0.f;
    for (int c = 0; c < NC; c++) { float d = r[c] - m; var += d * d; }
    var *= (1.f / NC);
    const float rs = rsqrtf(var + EPS);
    float* w = dst + tid * XLD;
    for (int c = 0; c < NC; c++) w[c] = (r[c] - m) * rs * g[c] + b[c];
  }
}
}  // namespace

__global__ void __launch_bounds__(128, 1)
slm_fwd_kernel(const int* __restrict__ idx, const int* __restrict__ targets,
               const float* __restrict__ tok_emb, const float* __restrict__ pos_emb,
               const float* __restrict__ ln1_g, const float* __restrict__ ln1_b,
               const float* __restrict__ wq, const float* __restrict__ wk,
               const float* __restrict__ wv, const float* __restrict__ wo,
               const float* __restrict__ bo, const float* __restrict__ ln2_g,
               const float* __restrict__ ln2_b, const float* __restrict__ w1,
               const float* __restrict__ b1, const float* __restrict__ w2,
               const float* __restrict__ b2, const float* __restrict__ lnf_g,
               const float* __restrict__ lnf_b, const float* __restrict__ lm_w,
               const float* __restrict__ lm_b,
               float* __restrict__ out_logits, float* __restrict__ block_loss) {
  const int b    = blockIdx.x;
  const int tid  = threadIdx.x;
  const int wave = tid >> 5;
  const int lane = tid & 31;
  const int half = lane >> 4;
  const int l16  = lane & 15;

  extern __shared__ float smem[];
  float* xs  = smem;                 // [64][68] residual stream
  float* hs  = xs + NT * XLD;        // [64][68] LN out / attn concat
  float* qs  = hs + NT * XLD;        // [4][64][17]
  float* ks2 = qs + NH * NT * QLD;   // [4][64][17]
  float* vs2 = ks2 + NH * NT * QLD;  // [4][64][17]
  float* sc  = vs2 + NH * NT * QLD;  // scores [4][64][68] / hidden [64][260] / logits [64][100]

  const v8f z8 = {0.f, 0.f, 0.f, 0.f, 0.f, 0.f, 0.f, 0.f};

  // token + positional embedding
  for (int i = tid; i < NT * NC; i += 128) {
    const int t = i >> 6, c = i & 63;
    xs[t * XLD + c] = tok_emb[(size_t)idx[(size_t)b * NT + t] * NC + c] + pos_emb[t * NC + c];
  }
  __syncthreads();

  for (int l = 0; l < NL; l++) {
    ln_rows(xs, hs, ln1_g + l * NC, ln1_b + l * NC, tid);
    __syncthreads();

    // ---- QKV: wave <-> head ----
    const int hd = wave;
    const float* wqh = wq + (size_t)(l * NH + hd) * NC * NHS;
    const float* wkh = wk + (size_t)(l * NH + hd) * NC * NHS;
    const float* wvh = wv + (size_t)(l * NH + hd) * NC * NHS;
    float* qh = qs + hd * NT * QLD;
    float* kh = ks2 + hd * NT * QLD;
    float* vh = vs2 + hd * NT * QLD;
    for (int mt = 0; mt < 4; mt++) {
      const float* Ar = hs + mt * 16 * XLD;
      store_tile(qh + mt * 16 * QLD, QLD, mm16(Ar, XLD, wqh, NHS, 1, NC, z8));
      store_tile(kh + mt * 16 * QLD, QLD, mm16(Ar, XLD, wkh, NHS, 1, NC, z8));
      store_tile(vh + mt * 16 * QLD, QLD, mm16(Ar, XLD, wvh, NHS, 1, NC, z8));
    }
    __syncthreads();

    // ---- scores = (q k^T) * HS^-0.5 ----
    float* W = sc + hd * NT * XLD;
    for (int mt = 0; mt < 4; mt++)
      for (int nt = 0; nt < 4; nt++) {
        v8f a = mm16(qh + mt * 16 * QLD, QLD, kh + nt * 16 * QLD, 1, QLD, NHS, z8);
        store_tile(W + mt * 16 * XLD + nt * 16, XLD, a * 0.25f);
      }
    __syncthreads();

    // ---- causal softmax: 2 rows per lane ----
    for (int rr = 0; rr < 2; rr++) {
      const int t = lane + rr * 32;
      float* row = W + t * XLD;
      float mx = -3.4e38f;
      for (int u = 0; u <= t; u++) mx = fmaxf(mx, row[u]);
      float s = 0.f;
      for (int u = 0; u <= t; u++) { const float e = __expf(row[u] - mx); row[u] = e; s += e; }
      const float inv = 1.f / s;
      for (int u = 0; u < NT; u++) row[u] = (u <= t) ? row[u] * inv : 0.f;
    }
    __syncthreads();

    // ---- o = wei @ v -> concat heads into hs ----
    for (int mt = 0; mt < 4; mt++)
      store_tile(hs + mt * 16 * XLD + hd * 16, XLD,
                 mm16(W + mt * 16 * XLD, XLD, vh, QLD, 1, NT, z8));
    __syncthreads();

    // ---- x += attn @ wo + bo : wave <-> row tile ----
    const float* wol = wo + (size_t)l * NC * NC;
    {
      const int mt = wave;
      for (int nt = 0; nt < 4; nt++) {
        v8f a = mm16(hs + mt * 16 * XLD, XLD, wol + nt * 16, NC, 1, NC, z8);
        const float bb = bo[l * NC + nt * 16 + l16];
        float* xrow = xs + mt * 16 * XLD + nt * 16;
#pragma unroll
        for (int i = 0; i < 8; i++) xrow[(i + 8 * half) * XLD + l16] += a[i] + bb;
      }
    }
    __syncthreads();

    ln_rows(xs, hs, ln2_g + l * NC, ln2_b + l * NC, tid);
    __syncthreads();

    // ---- hid = relu(h @ w1 + b1), streamed into sc ----
    const float* w1l = w1 + (size_t)l * NC * 4 * NC;
    {
      const int mt = wave;
      for (int jt = 0; jt < 16; jt++) {
        v8f a = mm16(hs + mt * 16 * XLD, XLD, w1l + jt * 16, 4 * NC, 1, NC, z8);
        const float bb = b1[l * 4 * NC + jt * 16 + l16];
        float* dst = sc + mt * 16 * HID_LD + jt * 16;
#pragma unroll
        for (int i = 0; i < 8; i++)
          dst[(i + 8 * half) * HID_LD + l16] = fmaxf(a[i] + bb, 0.f);
      }
    }
    __syncthreads();

    // ---- x += hid @ w2 + b2 ----
    const float* w2l = w2 + (size_t)l * 4 * NC * NC;
    {
      const int mt = wave;
      for (int nt = 0; nt < 4; nt++) {
        v8f a = mm16(sc + mt * 16 * HID_LD, HID_LD, w2l + nt * 16, NC, 1, 4 * NC, z8);
        const float bb = b2[l * NC + nt * 16 + l16];
        float* xrow = xs + mt * 16 * XLD + nt * 16;
#pragma unroll
        for (int i = 0; i < 8; i++) xrow[(i + 8 * half) * XLD + l16] += a[i] + bb;
      }
    }
    __syncthreads();
  }

  ln_rows(xs, hs, lnf_g, lnf_b, tid);
  __syncthreads();

  // ---- logits = h @ lm_w + lm_b -> global + LDS staging for loss ----
  {
    const int mt = wave;
    for (int nt = 0; nt < 6; nt++) {
      v8f a = mm16(hs + mt * 16 * XLD, XLD, lm_w + nt * 16, NV, 1, NC, z8);
      const float bb = lm_b[nt * 16 + l16];
      float* go = out_logits + ((size_t)b * NT + mt * 16) * NV + nt * 16;
      float* ls = sc + mt * 16 * LLD + nt * 16;
#pragma unroll
      for (int i = 0; i < 8; i++) {
        const float val = a[i] + bb;
        go[(size_t)(i + 8 * half) * NV + l16] = val;
        ls[(i + 8 * half) * LLD + l16] = val;
      }
    }
  }
  __syncthreads();

  // ---- per-row cross-entropy partials, fixed-order block sum ----
  float* red = qs;  // reuse
  if (tid < NT) {
    const float* row = sc + tid * LLD;
    float mx = -3.4e38f;
    for (int vv = 0; vv < NV; vv++) mx = fmaxf(mx, row[vv]);
    float s = 0.f;
    for (int vv = 0; vv < NV; vv++) s += __expf(row[vv] - mx);
    const float lse = __logf(s) + mx;
    const int tg = targets[(size_t)b * NT + tid];
    red[tid] = lse - row[tg];
  }
  __syncthreads();
  if (tid == 0) {
    float s = 0.f;
    for (int t = 0; t < NT; t++) s += red[t];
    block_loss[b] = s;
  }
}

__global__ void slm_loss_reduce(const float* __restrict__ partial,
                                float* __restrict__ loss_out) {
  __shared__ float sm[256];
  float s = 0.f;
  for (int i = threadIdx.x; i < NB; i += 256) s += partial[i];
  sm[threadIdx.x] = s;
  __syncthreads();
  for (int st = 128; st > 0; st >>= 1) {
    if ((int)threadIdx.x < st) sm[threadIdx.x] += sm[threadIdx.x + st];
    __syncthreads();
  }
  if (threadIdx.x == 0) *loss_out = sm[0] / (float)((size_t)NB * NT);
}

extern "C" void kernel_launch(void* const* d_in, const int* in_sizes, int n_in,
                              void* d_out, int out_size, void* d_ws, size_t ws_size,
                              hipStream_t stream) {
  const int*   idx     = (const int*)d_in[0];
  const int*   targets = (const int*)d_in[1];
  const float* tok_emb = (const float*)d_in[2];
  const float* pos_emb = (const float*)d_in[3];
  const float* ln1_g   = (const float*)d_in[4];
  const float* ln1_b   = (const float*)d_in[5];
  const float* wq      = (const float*)d_in[6];
  const float* wk      = (const float*)d_in[7];
  const float* wv      = (const float*)d_in[8];
  const float* wo      = (const float*)d_in[9];
  const float* bo      = (const float*)d_in[10];
  const float* ln2_g   = (const float*)d_in[11];
  const float* ln2_b   = (const float*)d_in[12];
  const float* w1      = (const float*)d_in[13];
  const float* b1      = (const float*)d_in[14];
  const float* w2      = (const float*)d_in[15];
  const float* b2      = (const float*)d_in[16];
  const float* lnf_g   = (const float*)d_in[17];
  const float* lnf_b   = (const float*)d_in[18];
  const float* lm_w    = (const float*)d_in[19];
  const float* lm_b    = (const float*)d_in[20];

  float* out_logits = (float*)d_out;
  float* loss_out   = out_logits + (size_t)NB * NT * NV;
  float* partial    = (float*)d_ws;  // NB floats

  constexpr size_t SMEM =
      (size_t)(2 * NT * XLD + 3 * NH * NT * QLD + NH * NT * XLD) * sizeof(float);  // ~153 KB

  slm_fwd_kernel<<<NB, 128, SMEM, stream>>>(idx, targets, tok_emb, pos_emb, ln1_g, ln1_b,
                                            wq, wk, wv, wo, bo, ln2_g, ln2_b, w1, b1, w2,
                                            b2, lnf_g, lnf_b, lm_w, lm_b, out_logits,
                                            partial);
  slm_loss_reduce<<<1, 256, 0, stream>>>(partial, loss_out);
}